// NeuralDecoder_88802743812478
// MI455X (gfx1250) — compile-verified
//
#include <hip/hip_runtime.h>
#include <hip/hip_bf16.h>
#include <stdint.h>

// ---------------------------------------------------------------------------
// Problem constants (match reference)
// ---------------------------------------------------------------------------
#define B         256
#define N_VARS    8192
#define N_CHECKS  4096
#define N_NODES   (N_VARS + N_CHECKS)          // 12288
#define N_EDGES   16777216
#define N_ITERS   10
#define NEG_SLOPE 0.01f

#define NODES_TOT ((size_t)B * N_NODES)        // 3,145,728
#define VARS_TOT  ((size_t)B * N_VARS)         // 2,097,152

// Edge kernel tiling
#define TPB       256                          // threads per block (8 waves)
#define EPT       4                            // edges per thread per tile (one b128)
#define TILE      (TPB * EPT)                  // 1024 edges / tile
#define EDGE_WGS  4096                         // workgroups for edge kernel
#define NT_TILES  (N_EDGES / TILE)             // 16384 tiles (exact, 4 per WG)

// native clang vector type (usable with __builtin_nontemporal_store)
typedef float fvec4 __attribute__((ext_vector_type(4)));

// ---------------------------------------------------------------------------
// CDNA5 async-copy helpers (gfx1250): global -> LDS DMA tracked by ASYNCcnt.
// Each lane supplies its own 32-bit LDS byte offset (wave-relative) and a
// 64-bit global address; one instruction moves 16B per lane.
// th:TH_LOAD_NT: the edge stream is one-touch (192 MB/iter ~= whole L2), so
// mark it non-temporal to keep the L2 reserved for the 25 MB node state that
// the random gathers + atomics hammer.
// ---------------------------------------------------------------------------
__device__ __forceinline__ uint32_t lds_off(const void* p) {
  // addrspace(3) -> generic keeps the LDS byte offset in the low 32 bits
  return (uint32_t)(size_t)p;
}

__device__ __forceinline__ void async_copy_b128_nt(uint32_t lds_byte_off,
                                                   const void* gaddr) {
  asm volatile("global_load_async_to_lds_b128 %0, %1, off th:TH_LOAD_NT"
               :
               : "v"(lds_byte_off),
                 "v"((unsigned long long)(size_t)gaddr)
               : "memory");
}

__device__ __forceinline__ void wait_async_le(int n) {
#if __has_builtin(__builtin_amdgcn_s_wait_asynccnt)
  if (n == 0) __builtin_amdgcn_s_wait_asynccnt(0);
  else        __builtin_amdgcn_s_wait_asynccnt(3);
#else
  if (n == 0) asm volatile("s_wait_asynccnt 0" ::: "memory");
  else        asm volatile("s_wait_asynccnt 3" ::: "memory");
#endif
}

// ---------------------------------------------------------------------------
// x = x0 = [initial_llrs | zeros]   (float4-vectorized; N_NODES % 4 == 0 and
// the 8192-column boundary is float4-aligned, so each vec4 is wholly in-range)
// ---------------------------------------------------------------------------
__global__ __launch_bounds__(TPB) void init_x_kernel(float* __restrict__ x,
                                                     const float* __restrict__ llrs) {
  size_t q = (size_t)blockIdx.x * blockDim.x + threadIdx.x;     // vec4 index
  if (q >= NODES_TOT / 4) return;
  size_t i = q * 4;
  int b = (int)(i / N_NODES);
  int c = (int)(i - (size_t)b * N_NODES);
  fvec4 v;
  if (c < N_VARS) {
    v = *reinterpret_cast<const fvec4*>(&llrs[(size_t)b * N_VARS + c]);
  } else {
    v = (fvec4)0.0f;
  }
  *reinterpret_cast<fvec4*>(&x[i]) = v;
}

// ---------------------------------------------------------------------------
// out[b, v] = x[b, v]  for v < N_VARS   (per-iteration output slab).
// Output is write-once / never re-read on device -> non-temporal stores.
// ---------------------------------------------------------------------------
__global__ __launch_bounds__(TPB) void copy_out_kernel(float* __restrict__ out,
                                                       const float* __restrict__ x) {
  size_t q = (size_t)blockIdx.x * blockDim.x + threadIdx.x;     // vec4 index
  if (q >= VARS_TOT / 4) return;
  size_t i = q * 4;
  int b = (int)(i >> 13);          // / 8192
  int v = (int)(i & 8191);
  const fvec4 val = *reinterpret_cast<const fvec4*>(&x[(size_t)b * N_NODES + v]);
  __builtin_nontemporal_store(val, reinterpret_cast<fvec4*>(&out[i]));
}

// ---------------------------------------------------------------------------
// Edge kernel: double-buffered async-to-LDS stream of (src, dst, mask),
// L2-resident gathers of node state, per-edge nonlinearity, f32 atomic scatter.
// LDS regions are partitioned by threadIdx.x and each thread consumes only the
// slice it DMA'd, so the only synchronization needed is the per-wave ASYNCcnt.
// ---------------------------------------------------------------------------
__global__ __launch_bounds__(TPB) void edge_kernel(
    const int*   __restrict__ src,
    const int*   __restrict__ dst,
    const float* __restrict__ msk,
    const float* __restrict__ xcur,
    float*       __restrict__ xnext,
    const float* __restrict__ attW,
    const float* __restrict__ attb,
    const float* __restrict__ scal,
    const float* __restrict__ pen,
    int it) {
  __shared__ __align__(16) int   s_s[2][TILE];
  __shared__ __align__(16) int   s_d[2][TILE];
  __shared__ __align__(16) float s_m[2][TILE];

  const int tid = threadIdx.x;
  // uniform per-iteration layer parameters (scalar loads)
  const float W0 = attW[it * 3 + 0];
  const float W1 = attW[it * 3 + 1];
  const float W2 = attW[it * 3 + 2];
  const float bb = attb[it];
  const float sc = scal[it];
  const float pn = pen[it];

  const int G = gridDim.x;

  auto issue = [&](int buf, int tile) {
    const int eb = tile * TILE + tid * EPT;         // <= 2^24, fits int
    async_copy_b128_nt(lds_off(&s_s[buf][tid * EPT]), src + eb);
    async_copy_b128_nt(lds_off(&s_d[buf][tid * EPT]), dst + eb);
    async_copy_b128_nt(lds_off(&s_m[buf][tid * EPT]), msk + eb);
  };

  int t = blockIdx.x;
  if (t < NT_TILES) issue(0, t);

  int buf = 0;
  for (; t < NT_TILES; t += G, buf ^= 1) {
    const int tn = t + G;
    const bool pf = (tn < NT_TILES);
    if (pf) issue(buf ^ 1, tn);                     // prefetch next tile
    wait_async_le(pf ? 3 : 0);                      // drain this tile's 3 DMAs

    // ds_load_b128 readback of this thread's slice
    const int4   s4 = *reinterpret_cast<const int4*>(&s_s[buf][tid * EPT]);
    const int4   d4 = *reinterpret_cast<const int4*>(&s_d[buf][tid * EPT]);
    const float4 m4 = *reinterpret_cast<const float4*>(&s_m[buf][tid * EPT]);

    // L2-resident random gathers (node state = 12.6 MB << 192 MB L2)
    const float xs0 = xcur[s4.x], xs1 = xcur[s4.y], xs2 = xcur[s4.z], xs3 = xcur[s4.w];
    const float xd0 = xcur[d4.x], xd1 = xcur[d4.y], xd2 = xcur[d4.z], xd3 = xcur[d4.w];

    #define EDGE_MSG(XS, XD, M, DI)                                            \
      {                                                                        \
        float raw = fmaf((XS), W0, fmaf((XD), W1, fmaf((M), W2, bb)));         \
        raw = (raw >= 0.f) ? raw : NEG_SLOPE * raw;                            \
        raw = fmaf((M), pn, raw);                                              \
        const float sig = 1.0f / (1.0f + __expf(-raw));                        \
        atomicAdd(&xnext[DI], (XS) * sig * sc);                                \
      }

    EDGE_MSG(xs0, xd0, m4.x, d4.x)
    EDGE_MSG(xs1, xd1, m4.y, d4.y)
    EDGE_MSG(xs2, xd2, m4.z, d4.z)
    EDGE_MSG(xs3, xd3, m4.w, d4.w)
    #undef EDGE_MSG
  }
}

// ---------------------------------------------------------------------------
// Host-side orchestration (graph-capture safe: launches only)
// ---------------------------------------------------------------------------
extern "C" void kernel_launch(void* const* d_in, const int* in_sizes, int n_in,
                              void* d_out, int out_size, void* d_ws, size_t ws_size,
                              hipStream_t stream) {
  (void)in_sizes; (void)n_in; (void)out_size; (void)ws_size;

  const float* llrs = (const float*)d_in[0];
  const int*   eidx = (const int*)  d_in[1];
  const float* msk  = (const float*)d_in[2];
  const float* attW = (const float*)d_in[3];
  const float* attb = (const float*)d_in[4];
  const float* scal = (const float*)d_in[5];
  const float* pen  = (const float*)d_in[6];
  float*       out  = (float*)d_out;

  const int* src = eidx;
  const int* dst = eidx + (size_t)N_EDGES;

  // ping-pong node state in workspace: 2 * 3,145,728 floats = 25.2 MB
  float* xa = (float*)d_ws;
  float* xb = xa + NODES_TOT;

  const int initBlocks = (int)((NODES_TOT / 4 + TPB - 1) / TPB);  // 3072
  const int copyBlocks = (int)((VARS_TOT  / 4 + TPB - 1) / TPB);  // 2048

  // x_cur = x0
  init_x_kernel<<<initBlocks, TPB, 0, stream>>>(xa, llrs);

  for (int it = 0; it < N_ITERS; ++it) {
    // x_next = x0 (residual base), then accumulate messages into it
    init_x_kernel<<<initBlocks, TPB, 0, stream>>>(xb, llrs);
    edge_kernel<<<EDGE_WGS, TPB, 0, stream>>>(src, dst, msk, xa, xb,
                                              attW, attb, scal, pen, it);
    copy_out_kernel<<<copyBlocks, TPB, 0, stream>>>(out + (size_t)it * VARS_TOT, xb);
    // swap roles
    float* tmp = xa; xa = xb; xb = tmp;
  }
}